// DGCNN_41618233099167
// MI455X (gfx1250) — compile-verified
//
#include <hip/hip_runtime.h>
#include <float.h>

#define KNN   20
#define CHMAX 128   // max per-layer input channels (layer3: 128)

typedef float v2f __attribute__((ext_vector_type(2)));
typedef float v8f __attribute__((ext_vector_type(8)));

static __device__ __forceinline__ v8f wmma4(v2f a, v2f b, v8f c) {
  // D = A(16x4 f32) * B(4x16 f32) + C(16x16 f32)
  return __builtin_amdgcn_wmma_f32_16x16x4_f32(false, a, false, b, (short)0, c,
                                               false, false);
}

// --- CDNA5 async global->LDS copy path (ASYNCcnt tracked) -------------------
static __device__ __forceinline__ unsigned lds_off(const void* p) {
  // ISA: LDS aperture flat address maps to LDS via addr[31:0].
  return (unsigned)(unsigned long long)p;
}
static __device__ __forceinline__ void async_ld_b32(unsigned loff, const void* g) {
  asm volatile("global_load_async_to_lds_b32 %0, %1, off"
               :: "v"(loff), "v"(g) : "memory");
}
static __device__ __forceinline__ void wait_async0() {
  asm volatile("s_wait_asynccnt 0" ::: "memory");
}

// ---------------------------------------------------------------------------
// Kernel 1: fused pairwise-distance (WMMA gram tiles) + per-row top-20 kNN.
// One wave handles 16 query rows; B tiles are double-buffered with async
// global->LDS copies overlapping the WMMA + top-k work on the current tile.
// ---------------------------------------------------------------------------
__global__ __launch_bounds__(32) void knn_topk_kernel(
    const float* __restrict__ X, int rowStride, int C, int N,
    int* __restrict__ idxOut)
{
  __shared__ float sA[16 * CHMAX];
  __shared__ float sB[2][16 * CHMAX];
  __shared__ float sqA[16];
  __shared__ float sqB[16];
  __shared__ float tile[16 * 16];
  __shared__ float topv[16 * KNN];
  __shared__ int   topi[16 * KNN];

  const int lane = threadIdx.x;
  const int tilesPerB = N / 16;
  const int b  = blockIdx.x / tilesPerB;
  const int m0 = (blockIdx.x % tilesPerB) * 16;
  const int Cpad = (C + 3) & ~3;

  // Zero pad-sensitive regions once; async stages only write c < C entries.
  for (int i = lane; i < 16 * Cpad; i += 32) {
    sA[i] = 0.0f; sB[0][i] = 0.0f; sB[1][i] = 0.0f;
  }
  for (int i = lane; i < 16 * KNN; i += 32) { topv[i] = -FLT_MAX; topi[i] = 0; }
  __syncthreads();   // zeros land before async writes

  // Async-stage the 16 query rows and the first B tile.
  for (int i = lane; i < 16 * C; i += 32) {
    int r = i / C, c = i % C;
    async_ld_b32(lds_off(&sA[r * Cpad + c]),
                 X + (size_t)(b * N + m0 + r) * rowStride + c);
  }
  for (int i = lane; i < 16 * C; i += 32) {
    int r = i / C, c = i % C;
    async_ld_b32(lds_off(&sB[0][r * Cpad + c]),
                 X + (size_t)(b * N + r) * rowStride + c);
  }
  wait_async0();
  __syncthreads();

  if (lane < 16) {
    float s = 0.f;
    for (int c = 0; c < C; ++c) { float v = sA[lane * Cpad + c]; s += v * v; }
    sqA[lane] = s;
  }
  __syncthreads();

  const int hf  = lane >> 4;   // half-wave id: selects K-slice {0,1} vs {2,3}
  const int l16 = lane & 15;

  int cur = 0;
  for (int n0 = 0; n0 < N; n0 += 16) {
    // Kick async copy of the NEXT tile into the other buffer.
    if (n0 + 16 < N) {
      const int nxt = cur ^ 1;
      for (int i = lane; i < 16 * C; i += 32) {
        int r = i / C, c = i % C;
        async_ld_b32(lds_off(&sB[nxt][r * Cpad + c]),
                     X + (size_t)(b * N + n0 + 16 + r) * rowStride + c);
      }
    }

    if (lane < 16) {
      float s = 0.f;
      for (int c = 0; c < C; ++c) { float v = sB[cur][lane * Cpad + c]; s += v * v; }
      sqB[lane] = s;
    }
    __syncthreads();

    // Gram tile via f32 WMMA: rows = queries m0..m0+15, cols = points n0..n0+15
    v8f acc = {};
    for (int c0 = 0; c0 < Cpad; c0 += 4) {
      int kb = c0 + 2 * hf;
      v2f a, bv;
      a.x  = sA[l16 * Cpad + kb];
      a.y  = sA[l16 * Cpad + kb + 1];
      bv.x = sB[cur][l16 * Cpad + kb];
      bv.y = sB[cur][l16 * Cpad + kb + 1];
      acc = wmma4(a, bv, acc);
    }
    // neg_d = 2*dot - |xi|^2 - |xj|^2 ; D layout: lane=col l16, vgpr r -> row hf*8+r
    float sqn = sqB[l16];
    #pragma unroll
    for (int r = 0; r < 8; ++r) {
      int m = hf * 8 + r;
      tile[m * 16 + l16] = 2.0f * acc[r] - sqA[m] - sqn;
    }
    __syncthreads();

    // Per-row sorted top-20 update (stable: ties keep lowest index first).
    if (lane < 16) {
      float* tv = &topv[lane * KNN];
      int*   ti = &topi[lane * KNN];
      for (int j = 0; j < 16; ++j) {
        float v = tile[lane * 16 + j];
        if (v > tv[KNN - 1]) {
          int p = KNN - 1;
          while (p > 0 && tv[p - 1] < v) {
            tv[p] = tv[p - 1]; ti[p] = ti[p - 1]; --p;
          }
          tv[p] = v; ti[p] = n0 + j;
        }
      }
    }
    wait_async0();      // next tile resident before buffer swap
    __syncthreads();
    cur ^= 1;
  }

  for (int i = lane; i < 16 * KNN; i += 32) {
    int r = i / KNN, kk = i % KNN;
    idxOut[(size_t)(b * N + m0 + r) * KNN + kk] = topi[r * KNN + kk];
  }
}

// Edge feature element: [x_j - x_i ; x_i] with zero padding to cpad.
static __device__ __forceinline__ float ef(const float* sNbr, const float* sCtr,
                                           int kk, int c, int ch, int cin2) {
  bool lo = (c < ch);
  int cc = lo ? c : (c - ch);
  cc = (cc < ch) ? cc : 0;                 // clamp padded-K reads
  float ctr = sCtr[cc];
  float nb  = sNbr[kk * ch + cc];
  float v = lo ? (nb - ctr) : ctr;
  return (c < cin2) ? v : 0.0f;
}

// ---------------------------------------------------------------------------
// Kernel 2: edge conv. One wave = one (point, 16-output tile).
// Neighbor/center features staged with async global->LDS copies.
// 20 edges -> two 16-row WMMA M-tiles (rows 20..31 padded & excluded from max);
// max over k done in-register from the D layout + one cross-half shuffle.
// ---------------------------------------------------------------------------
__global__ __launch_bounds__(32) void edgeconv_kernel(
    const float* __restrict__ X, int rowStride, int ch,
    const int* __restrict__ idx,
    const float* __restrict__ W, const float* __restrict__ bias,
    float* __restrict__ Fout, int outStride, int outOff,
    int Co, int N)
{
  __shared__ float sCtr[CHMAX];
  __shared__ float sNbr[KNN * CHMAX];
  __shared__ int   sIdx[KNN];

  const int lane = threadIdx.x;
  const int oTiles = Co / 16;
  const int ot = blockIdx.x % oTiles;
  const int p  = blockIdx.x / oTiles;
  const int n  = p % N;
  const int b  = p / N;
  const int o0 = ot * 16;

  const int cin2 = 2 * ch;
  const int cpad = (cin2 + 3) & ~3;

  if (lane < KNN) sIdx[lane] = idx[(size_t)(b * N + n) * KNN + lane];
  __syncthreads();

  for (int c = lane; c < ch; c += 32)
    async_ld_b32(lds_off(&sCtr[c]), X + (size_t)(b * N + n) * rowStride + c);
  for (int i = lane; i < KNN * ch; i += 32) {
    int kk = i / ch, c = i % ch;
    async_ld_b32(lds_off(&sNbr[i]),
                 X + (size_t)(b * N + sIdx[kk]) * rowStride + c);
  }
  wait_async0();
  __syncthreads();

  const int hf  = lane >> 4;
  const int l16 = lane & 15;
  const int oc  = o0 + l16;

  v8f acc0 = {}, acc1 = {};
  for (int c0 = 0; c0 < cpad; c0 += 4) {
    int kb = c0 + 2 * hf;
    v2f bv;
    bv.x = (kb     < cin2) ? W[(size_t)oc * cin2 + kb]     : 0.0f;
    bv.y = (kb + 1 < cin2) ? W[(size_t)oc * cin2 + kb + 1] : 0.0f;
    v2f a0, a1;
    a0.x = ef(sNbr, sCtr, l16, kb,     ch, cin2);
    a0.y = ef(sNbr, sCtr, l16, kb + 1, ch, cin2);
    int  kk1  = 16 + l16;
    bool v1   = kk1 < KNN;
    int  kk1c = v1 ? kk1 : 0;
    a1.x = v1 ? ef(sNbr, sCtr, kk1c, kb,     ch, cin2) : 0.0f;
    a1.y = v1 ? ef(sNbr, sCtr, kk1c, kb + 1, ch, cin2) : 0.0f;
    acc0 = wmma4(a0, bv, acc0);
    acc1 = wmma4(a1, bv, acc1);
  }

  const float bb = bias[oc];
  float pm = -FLT_MAX;
  #pragma unroll
  for (int r = 0; r < 8; ++r) {          // tile0 rows: k = 0..15, all valid
    float v = acc0[r] + bb;
    v = (v > 0.f) ? v : 0.2f * v;
    pm = fmaxf(pm, v);
  }
  #pragma unroll
  for (int r = 0; r < 4; ++r) {          // tile1: only k = 16..19 (half 0, rows 0..3)
    float v = acc1[r] + bb;
    v = (v > 0.f) ? v : 0.2f * v;
    v = (hf == 0) ? v : -FLT_MAX;
    pm = fmaxf(pm, v);
  }
  pm = fmaxf(pm, __shfl_xor(pm, 16, 32));
  if (lane < 16)
    Fout[(size_t)(b * N + n) * outStride + outOff + oc] = pm;
}

// ---------------------------------------------------------------------------
// Kernel 3: final 512->1024 GEMM fused with global max over N.
// One wave = one (batch, 16-output tile); its 32KB Wf slice is staged once
// into LDS (async) and reused across all 128 point tiles.
// ---------------------------------------------------------------------------
__global__ __launch_bounds__(32) void final_kernel(
    const float* __restrict__ F, const float* __restrict__ Wf,
    const float* __restrict__ bf, float* __restrict__ out,
    int N, int CF, int CO)
{
  __shared__ float sW[16 * 512];   // CF == 512

  const int lane = threadIdx.x;
  const int oTiles = CO / 16;
  const int ot = blockIdx.x % oTiles;
  const int b  = blockIdx.x / oTiles;
  const int o0 = ot * 16;
  const int hf  = lane >> 4;
  const int l16 = lane & 15;
  const int oc  = o0 + l16;
  const float bb = bf[oc];

  for (int i = lane; i < 16 * CF; i += 32) {
    int r = i / CF, c = i % CF;
    async_ld_b32(lds_off(&sW[i]), Wf + (size_t)(o0 + r) * CF + c);
  }
  wait_async0();
  __syncthreads();

  float pm = -FLT_MAX;
  for (int n0 = 0; n0 < N; n0 += 16) {
    if (n0 + 16 < N)   // warm next A tile row for this lane
      __builtin_prefetch(F + (size_t)(b * N + n0 + 16 + l16) * CF, 0, 0);
    v8f acc = {};
    for (int c0 = 0; c0 < CF; c0 += 4) {
      int kb = c0 + 2 * hf;
      v2f a, bv;
      a.x  = F[(size_t)(b * N + n0 + l16) * CF + kb];
      a.y  = F[(size_t)(b * N + n0 + l16) * CF + kb + 1];
      bv.x = sW[l16 * CF + kb];
      bv.y = sW[l16 * CF + kb + 1];
      acc = wmma4(a, bv, acc);
    }
    #pragma unroll
    for (int r = 0; r < 8; ++r) pm = fmaxf(pm, acc[r] + bb);
  }
  pm = fmaxf(pm, __shfl_xor(pm, 16, 32));
  if (lane < 16) out[(size_t)b * CO + oc] = pm;   // every element written once
}

// ---------------------------------------------------------------------------
extern "C" void kernel_launch(void* const* d_in, const int* in_sizes, int n_in,
                              void* d_out, int out_size, void* d_ws, size_t ws_size,
                              hipStream_t stream) {
  (void)in_sizes; (void)n_in; (void)out_size; (void)ws_size;
  const int B = 8, N = 2048, CF = 512, CO = 1024;

  const float* x  = (const float*)d_in[0];
  const float* W[4]  = {(const float*)d_in[1], (const float*)d_in[3],
                        (const float*)d_in[5], (const float*)d_in[7]};
  const float* bi[4] = {(const float*)d_in[2], (const float*)d_in[4],
                        (const float*)d_in[6], (const float*)d_in[8]};
  const float* Wf = (const float*)d_in[9];
  const float* bf = (const float*)d_in[10];
  float* out = (float*)d_out;

  float* F      = (float*)d_ws;                                   // (B,N,512) concat feats
  int*   idxBuf = (int*)((char*)d_ws + (size_t)B * N * CF * 4);   // (B,N,20)

  const int chIn[4]   = {3, 64, 64, 128};
  const int Co[4]     = {64, 64, 128, 256};
  const int inOff[4]  = {0, 0, 64, 128};    // prev layer's slice in F
  const int outOff[4] = {0, 64, 128, 256};  // this layer's slice in F

  for (int L = 0; L < 4; ++L) {
    const float* Xin = (L == 0) ? x : (F + inOff[L]);
    int stride = (L == 0) ? 3 : CF;
    knn_topk_kernel<<<dim3(B * (N / 16)), dim3(32), 0, stream>>>(
        Xin, stride, chIn[L], N, idxBuf);
    edgeconv_kernel<<<dim3(B * N * (Co[L] / 16)), dim3(32), 0, stream>>>(
        Xin, stride, chIn[L], idxBuf, W[L], bi[L], F, CF, outOff[L], Co[L], N);
  }
  final_kernel<<<dim3(B * (CO / 16)), dim3(32), 0, stream>>>(
      F, Wf, bf, out, N, CF, CO);
}